// FreTransfer_25864293056755
// MI455X (gfx1250) — compile-verified
//
#include <hip/hip_runtime.h>
#include <cstdint>

// ---------------------------------------------------------------------------
// Haar DWT, one level, stride-2, no padding.
//   in : x[8][16][512][512] fp32
//   out: out[8][64][256][256] fp32, channel = f*16 + c, f in {h00,h01,h10,h11}
//
// Memory-bound (268 MB, <1 FLOP/byte -> ~11.5 us at 23.3 TB/s). Structure:
// per-wave32 autonomous async pipeline. Each wave stages its own 4 KB input
// row pair via 8x GLOBAL_LOAD_ASYNC_TO_LDS_B128 (ASYNCcnt-tracked, private
// LDS double buffer), no workgroup barriers at all, b128 LDS reads,
// non-temporal b64 stores (256 B per wave per store instruction).
// ---------------------------------------------------------------------------

typedef float v2f __attribute__((ext_vector_type(2)));
typedef float v4f __attribute__((ext_vector_type(4)));

#define WAVE_TASKS 4     // row-pair tasks per wave (divides 256 -> same plane)
#define WPB        4     // waves per block (block = 128 threads)
#define NBLK       2048  // 32768 tasks / (WPB * WAVE_TASKS)

__global__ __launch_bounds__(128) void haar_dwt_wavepipe_kernel(
    const float* __restrict__ x,
    const float* __restrict__ h00, const float* __restrict__ h01,
    const float* __restrict__ h10, const float* __restrict__ h11,
    float* __restrict__ out)
{
    // 4 waves x (2 x 4 KB) private double buffers = 32 KB
    __shared__ float buf[WPB * 2048];

    const uint32_t l = threadIdx.x & 31u;   // lane in wave
    const uint32_t w = threadIdx.x >> 5;    // wave in block

    // Filter coefficients, layout [p][q]: {p0q0, p0q1, p1q0, p1q1}.
    const v4f f0 = *(const v4f*)h00;
    const v4f f1 = *(const v4f*)h01;
    const v4f f2 = *(const v4f*)h10;
    const v4f f3 = *(const v4f*)h11;

    // This wave's task stream: WAVE_TASKS consecutive (b,c,h) row pairs,
    // guaranteed inside one (b,c) plane since WAVE_TASKS divides 256.
    const uint32_t wv    = blockIdx.x * WPB + w;
    const uint32_t task0 = wv * WAVE_TASKS;
    const uint32_t plane = task0 >> 8;        // b*16 + c
    const uint32_t h0    = task0 & 255u;
    const uint32_t bb    = plane >> 4;
    const uint32_t cc    = plane & 15u;

    const uint64_t src       = (uint64_t)(uintptr_t)x;
    const uint32_t planeByte = plane * (512u * 512u * 4u);

    // LDS aperture: low 32 bits of the flat address == LDS byte offset.
    const uint32_t ldsW = (uint32_t)(uintptr_t)&buf[0] + w * 8192u;

    // Stage task `it` (rows 2(h0+it), 2(h0+it)+1 = 4 KB) into this wave's
    // buffer `bsel`: 8 async b128 issues, each moving a contiguous 512 B.
    #define ISSUE_TASK(it, bsel)                                                   \
      { _Pragma("unroll")                                                          \
        for (uint32_t k = 0; k < 8u; ++k) {                                        \
          uint32_t lo = ldsW + (uint32_t)(bsel) * 4096u + 16u * (l + 32u * k);     \
          int32_t  go = (int32_t)(planeByte + (h0 + (uint32_t)(it)) * 4096u        \
                                  + 16u * (l + 32u * k));                          \
          asm volatile("global_load_async_to_lds_b128 %0, %1, %2"                  \
                       :: "v"(lo), "v"(go), "s"(src) : "memory");                  \
        } }

    ISSUE_TASK(0, 0);

    const uint32_t outBase = (bb * 64u + cc) * 65536u;   // filter stride = 1048576

    #pragma unroll
    for (int it = 0; it < WAVE_TASKS; ++it) {
        if (it + 1 < WAVE_TASKS) {
            // Buffer (it+1)&1 was read by *this wave* last iteration; DS reads
            // and async LDS writes use separate counters, so close the WAR
            // window before overwriting (effectively free: data already used).
            asm volatile("s_wait_dscnt 0x0" ::: "memory");
            ISSUE_TASK(it + 1, (it + 1) & 1);
            // In-order async completion: <=8 outstanding => task `it` landed.
            asm volatile("s_wait_asynccnt 0x8" ::: "memory");
        } else {
            asm volatile("s_wait_asynccnt 0x0" ::: "memory");
        }

        // Consume own buffer: per lane 4x two rows of float4 (2 blocks each).
        // b128 LDS reads at byte 16*(l+32k): benign 2-way bank split only.
        const v4f* b4 = (const v4f*)&buf[w * 2048u + ((uint32_t)it & 1u) * 1024u];
        const uint32_t ob = outBase + (h0 + (uint32_t)it) * 256u;

        #pragma unroll
        for (uint32_t k = 0; k < 4u; ++k) {
            const uint32_t j  = l + 32u * k;     // float4 index within row
            const v4f r0 = b4[j];                // row 2h : cols 4j..4j+3
            const v4f r1 = b4[128u + j];         // row 2h+1

            // Two 2x2 blocks: (x,y | x,y) and (z,w | z,w)
            v2f o0, o1, o2, o3;
            o0.x = fmaf(f0.w, r1.y, fmaf(f0.z, r1.x, fmaf(f0.y, r0.y, f0.x * r0.x)));
            o0.y = fmaf(f0.w, r1.w, fmaf(f0.z, r1.z, fmaf(f0.y, r0.w, f0.x * r0.z)));
            o1.x = fmaf(f1.w, r1.y, fmaf(f1.z, r1.x, fmaf(f1.y, r0.y, f1.x * r0.x)));
            o1.y = fmaf(f1.w, r1.w, fmaf(f1.z, r1.z, fmaf(f1.y, r0.w, f1.x * r0.z)));
            o2.x = fmaf(f2.w, r1.y, fmaf(f2.z, r1.x, fmaf(f2.y, r0.y, f2.x * r0.x)));
            o2.y = fmaf(f2.w, r1.w, fmaf(f2.z, r1.z, fmaf(f2.y, r0.w, f2.x * r0.z)));
            o3.x = fmaf(f3.w, r1.y, fmaf(f3.z, r1.x, fmaf(f3.y, r0.y, f3.x * r0.x)));
            o3.y = fmaf(f3.w, r1.w, fmaf(f3.z, r1.z, fmaf(f3.y, r0.w, f3.x * r0.z)));

            // Output pixels 2j, 2j+1 of each filter row: b64 NT stores,
            // 256 B contiguous per wave per instruction, never re-read.
            const uint32_t p = ob + 2u * j;
            __builtin_nontemporal_store(o0, (v2f*)(out + p));
            __builtin_nontemporal_store(o1, (v2f*)(out + p + 1048576u));
            __builtin_nontemporal_store(o2, (v2f*)(out + p + 2097152u));
            __builtin_nontemporal_store(o3, (v2f*)(out + p + 3145728u));
        }
    }
    #undef ISSUE_TASK
}

extern "C" void kernel_launch(void* const* d_in, const int* in_sizes, int n_in,
                              void* d_out, int out_size, void* d_ws, size_t ws_size,
                              hipStream_t stream) {
    (void)in_sizes; (void)n_in; (void)d_ws; (void)ws_size; (void)out_size;
    const float* x   = (const float*)d_in[0];
    const float* h00 = (const float*)d_in[1];
    const float* h01 = (const float*)d_in[2];
    const float* h10 = (const float*)d_in[3];
    const float* h11 = (const float*)d_in[4];
    float* out = (float*)d_out;

    haar_dwt_wavepipe_kernel<<<dim3(NBLK), dim3(128), 0, stream>>>(
        x, h00, h01, h10, h11, out);
}